// CNNModel_29274497089615
// MI455X (gfx1250) — compile-verified
//
#include <hip/hip_runtime.h>
#include <cmath>

typedef float v2f __attribute__((ext_vector_type(2)));
typedef float v8f __attribute__((ext_vector_type(8)));

#define ELEMN  1048576
#define HALFN  524288
#define WIDTHN 100
#define KSN    15
#define NCONV  17

// ---------------- dense1: h[100] = W1[100 x HALF] @ x[:HALF] ----------------
// Grid: 7 row-tiles x 512 column-chunks. Each block: 256 thr = 8 waves.
// Each wave: 16 rows x 128 columns via 32 x V_WMMA_F32_16X16X4_F32,
// B = x broadcast into all 16 N columns (all D columns identical).
// Rows 100..111 are clamped to row 99 (safe loads); their junk results land in
// partials rows >= 100 which the reducer never reads -> no guards in the hot loop.
#define D1_ROWT   7
#define D1_NCHUNK 512
#define D1_CHUNK  (HALFN / D1_NCHUNK)   // 1024 columns per block
#define D1_WCOLS  (D1_CHUNK / 8)        // 128 columns per wave
#define D1_PROWS  112                   // padded rows (7*16)

__global__ __launch_bounds__(256) void k_dense1(const float* __restrict__ W1,
                                                const float* __restrict__ x,
                                                float* __restrict__ partials)
{
    const int rt   = blockIdx.x / D1_NCHUNK;
    const int ch   = blockIdx.x % D1_NCHUNK;
    const int tid  = threadIdx.x;
    const int w    = tid >> 5;
    const int lane = tid & 31;
    const int m    = lane & 15;
    const int kh   = lane >> 4;

    const int row  = rt * 16 + m;
    const int rowc = row < WIDTHN ? row : (WIDTHN - 1);   // clamp, no zero-fill needed

    const int k0 = ch * D1_CHUNK + w * D1_WCOLS;
    const float* __restrict__ pa = W1 + (size_t)rowc * HALFN + k0 + 2 * kh;
    const float* __restrict__ pb = x + k0 + 2 * kh;

    v8f c;
#pragma unroll
    for (int j = 0; j < 8; ++j) c[j] = 0.0f;

#pragma unroll 4
    for (int it = 0; it < D1_WCOLS / 4; ++it) {
        const v2f a = *reinterpret_cast<const v2f*>(pa);   // global_load_b64
        const v2f b = *reinterpret_cast<const v2f*>(pb);   // global_load_b64
        c = __builtin_amdgcn_wmma_f32_16x16x4_f32(false, a, false, b,
                                                  (short)0, c, false, false);
        pa += 4;
        pb += 4;
    }

    __shared__ float warr[8][16];
    if (lane == 0) {
#pragma unroll
        for (int j = 0; j < 8; ++j) warr[w][j] = c[j];       // rows rt*16 + 0..7
    }
    if (lane == 16) {
#pragma unroll
        for (int j = 0; j < 8; ++j) warr[w][8 + j] = c[j];   // rows rt*16 + 8..15
    }
    __syncthreads();
    if (tid < 16) {
        float s = 0.0f;
#pragma unroll
        for (int w2 = 0; w2 < 8; ++w2) s += warr[w2][tid];   // fixed order
        partials[(size_t)ch * D1_PROWS + rt * 16 + tid] = s;
    }
}

// ---------------- reduce partials (fixed order) + b1 ----------------
__global__ void k_reduce1(const float* __restrict__ partials,
                          const float* __restrict__ b1,
                          float* __restrict__ h0)
{
    const int r = threadIdx.x;
    if (r < WIDTHN) {
        float s = b1[r];
        for (int ch = 0; ch < D1_NCHUNK; ++ch)
            s += partials[(size_t)ch * D1_PROWS + r];
        h0[r] = s;
    }
}

// ---------------- conv chain: 17 x (1ch Conv1d k=15 'same' + CELU) ----------------
__global__ void k_conv(const float* __restrict__ h0,
                       const float* __restrict__ conv_w,
                       const float* __restrict__ conv_b,
                       float* __restrict__ h1)
{
    __shared__ float sh[WIDTHN + 2 * (KSN / 2)];   // 114, 7-zero halo each side
    const int tid = threadIdx.x;
    if (tid < WIDTHN + 14)
        sh[tid] = (tid >= 7 && tid < WIDTHN + 7) ? h0[tid - 7] : 0.0f;
    __syncthreads();

    for (int l = 0; l < NCONV; ++l) {
        float acc = 0.0f;
        if (tid < WIDTHN) {
            acc = conv_b[l];
#pragma unroll
            for (int t = 0; t < KSN; ++t)
                acc += sh[tid + t] * conv_w[l * KSN + t];   // cross-correlation
            acc = acc > 0.0f ? acc : expm1f(acc);           // CELU alpha=1
        }
        __syncthreads();
        if (tid < WIDTHN) sh[tid + 7] = acc;
        __syncthreads();
    }
    if (tid < WIDTHN) h1[tid] = sh[tid + 7];
}

// ---------------- dense3: g[i] = relu(W3[i,:] @ h + b3[i]) ----------------
// Each block: 256 thr = 8 waves, 128 output rows. K = 100 = 25 WMMA steps.
__global__ __launch_bounds__(256) void k_dense3(const float* __restrict__ W3,
                                                const float* __restrict__ h,
                                                const float* __restrict__ b3,
                                                float* __restrict__ g)
{
    __shared__ float hs[WIDTHN];
    const int tid = threadIdx.x;
    if (tid < WIDTHN) hs[tid] = h[tid];
    __syncthreads();

    const int w    = tid >> 5;
    const int lane = tid & 31;
    const int m    = lane & 15;
    const int kh   = lane >> 4;

    const size_t base = (size_t)blockIdx.x * 128 + (size_t)w * 16;
    const float* __restrict__ pa = W3 + (base + m) * WIDTHN + 2 * kh;
    const float* __restrict__ pb = hs + 2 * kh;

    v8f c;
#pragma unroll
    for (int j = 0; j < 8; ++j) c[j] = 0.0f;

#pragma unroll
    for (int k = 0; k < WIDTHN; k += 4) {
        const v2f a = *reinterpret_cast<const v2f*>(pa);   // global_load_b64
        const v2f b = *reinterpret_cast<const v2f*>(pb);   // ds_load_b64
        c = __builtin_amdgcn_wmma_f32_16x16x4_f32(false, a, false, b,
                                                  (short)0, c, false, false);
        pa += 4;
        pb += 4;
    }

    if (lane == 0) {
#pragma unroll
        for (int j = 0; j < 8; ++j) {
            const size_t r2 = base + j;
            g[r2] = fmaxf(c[j] + b3[r2], 0.0f);
        }
    }
    if (lane == 16) {
#pragma unroll
        for (int j = 0; j < 8; ++j) {
            const size_t r2 = base + 8 + j;
            g[r2] = fmaxf(c[j] + b3[r2], 0.0f);
        }
    }
}

// ---------------- hierarchical inclusive scan over g[HALF] ----------------
#define SC_BLK 2048                  // elems per block (256 thr * 8)
#define SC_NB  (HALFN / SC_BLK)      // 256 blocks

__global__ __launch_bounds__(256) void k_scan_block(float* __restrict__ g,
                                                    float* __restrict__ chunkSums)
{
    __shared__ float ls[256];
    const int tid = threadIdx.x;
    const size_t base = (size_t)blockIdx.x * SC_BLK + (size_t)tid * 8;

    float v[8];
#pragma unroll
    for (int i = 0; i < 8; ++i) v[i] = g[base + i];
#pragma unroll
    for (int i = 1; i < 8; ++i) v[i] += v[i - 1];   // sequential inclusive
    const float ts = v[7];

    ls[tid] = ts;                                    // Hillis-Steele, fixed order
    for (int off = 1; off < 256; off <<= 1) {
        __syncthreads();
        const float t = (tid >= off) ? ls[tid - off] : 0.0f;
        __syncthreads();
        ls[tid] += t;
    }
    __syncthreads();
    const float excl = tid ? ls[tid - 1] : 0.0f;
#pragma unroll
    for (int i = 0; i < 8; ++i) g[base + i] = v[i] + excl;
    if (tid == 255) chunkSums[blockIdx.x] = ls[255];
}

__global__ __launch_bounds__(256) void k_scan_tops(const float* __restrict__ chunkSums,
                                                   float* __restrict__ offsets,
                                                   float* __restrict__ scal)
{
    __shared__ float ls[256];
    const int tid = threadIdx.x;
    ls[tid] = chunkSums[tid];
    for (int off = 1; off < 256; off <<= 1) {
        __syncthreads();
        const float t = (tid >= off) ? ls[tid - off] : 0.0f;
        __syncthreads();
        ls[tid] += t;
    }
    __syncthreads();
    offsets[tid] = tid ? ls[tid - 1] : 0.0f;
    if (tid == 255) scal[0] = ls[255];   // S = max of cumsum (relu => nondecreasing)
}

// ---------------- exp(cs - S) into out[0..HALF), per-block exp sums ----------------
__global__ __launch_bounds__(256) void k_expsum(const float* __restrict__ g,
                                                const float* __restrict__ offsets,
                                                const float* __restrict__ scal,
                                                float* __restrict__ out,
                                                float* __restrict__ expPart)
{
    __shared__ float red[256];
    const int tid = threadIdx.x;
    const int b   = blockIdx.x;
    const float off = offsets[b];
    const float S   = scal[0];
    const size_t base = (size_t)b * SC_BLK + (size_t)tid * 8;

    float s = 0.0f;
#pragma unroll
    for (int i = 0; i < 8; ++i) {
        const float e = expf(g[base + i] + off - S);
        out[base + i] = e;
        s += e;
    }
    red[tid] = s;
    __syncthreads();
    for (int st = 128; st > 0; st >>= 1) {          // fixed-order tree
        if (tid < st) red[tid] += red[tid + st];
        __syncthreads();
    }
    if (tid == 0) expPart[b] = red[0];
}

// ---------------- normalize lower half, mirror into upper half ----------------
__global__ __launch_bounds__(256) void k_finalize(float* __restrict__ out,
                                                  const float* __restrict__ expPart)
{
    __shared__ float dsh;
    const int tid = threadIdx.x;
    if (tid == 0) {
        float d = 0.0f;
        for (int j = 0; j < SC_NB; ++j) d += expPart[j];   // fixed order
        dsh = 2.0f * d;                                    // mirrored half doubles sum
    }
    __syncthreads();
    const float inv = 1.0f / dsh;
    const size_t base = (size_t)blockIdx.x * SC_BLK + (size_t)tid * 8;
#pragma unroll
    for (int i = 0; i < 8; ++i) {
        const size_t idx = base + i;
        const float v = out[idx] * inv;
        out[idx] = v;
        out[(size_t)ELEMN - 1 - idx] = v;
    }
}

// ---------------- launcher ----------------
extern "C" void kernel_launch(void* const* d_in, const int* in_sizes, int n_in,
                              void* d_out, int out_size, void* d_ws, size_t ws_size,
                              hipStream_t stream)
{
    const float* x      = (const float*)d_in[0];
    const float* W1     = (const float*)d_in[1];
    const float* b1     = (const float*)d_in[2];
    const float* conv_w = (const float*)d_in[3];
    const float* conv_b = (const float*)d_in[4];
    const float* W3     = (const float*)d_in[5];
    const float* b3     = (const float*)d_in[6];
    // d_in[7] = bias (1e-30): softmax is shift-invariant -> no-op.

    float* out = (float*)d_out;
    float* ws  = (float*)d_ws;

    // workspace layout (floats)
    float* partials  = ws;                                  // 512*112 = 57344
    float* h0        = ws + (size_t)D1_NCHUNK * D1_PROWS;   // 112
    float* h1        = h0 + D1_PROWS;                       // 112
    float* g         = h1 + D1_PROWS;                       // HALF (scanned in place)
    float* chunkSums = g + HALFN;                           // 256
    float* offsets   = chunkSums + SC_NB;                   // 256
    float* expPart   = offsets + SC_NB;                     // 256
    float* scal      = expPart + SC_NB;                     // 1

    k_dense1<<<dim3(D1_ROWT * D1_NCHUNK), dim3(256), 0, stream>>>(W1, x, partials);
    k_reduce1<<<dim3(1), dim3(128), 0, stream>>>(partials, b1, h0);
    k_conv<<<dim3(1), dim3(128), 0, stream>>>(h0, conv_w, conv_b, h1);
    k_dense3<<<dim3(HALFN / 128), dim3(256), 0, stream>>>(W3, h1, b3, g);
    k_scan_block<<<dim3(SC_NB), dim3(256), 0, stream>>>(g, chunkSums);
    k_scan_tops<<<dim3(1), dim3(256), 0, stream>>>(chunkSums, offsets, scal);
    k_expsum<<<dim3(SC_NB), dim3(256), 0, stream>>>(g, offsets, scal, out, expPart);
    k_finalize<<<dim3(SC_NB), dim3(256), 0, stream>>>(out, expPart);
}